// Att_Bilinear_layer2_keycat_textual_visual_11879879544292
// MI455X (gfx1250) — compile-verified
//
#include <hip/hip_runtime.h>

// Problem constants (from the reference): B=32, N=64, A=32, O=32, D=512, QD=512
#define BB 32
#define NN 64
#define AA 32
#define OO 32
#define DD 512
#define QQD 512
#define MM (BB * NN)   // 2048 rows for the inter GEMM

typedef __attribute__((ext_vector_type(2))) float v2f;
typedef __attribute__((ext_vector_type(4))) float v4f;
typedef __attribute__((ext_vector_type(8))) float v8f;

// ---------------------------------------------------------------------------
// Kernel 1: inter[m, d] = sum_k q[m, k] * W[k, d]
//   M = 2048, K = 512, N = 512, all fp32 -> V_WMMA_F32_16X16X4_F32.
// Grid: 128 workgroups (one per 16-row strip), 256 threads = 8 waves.
// Wave w owns the 16x64 column strip [w*64, w*64+64): 4 accumulators.
// Fragment layout (ISA 7.12.2, 32-bit 16x4 A / 4x16 B):
//   lanes 0-15  hold K = k0+0 (VGPR0), k0+1 (VGPR1), row/col = lane
//   lanes 16-31 hold K = k0+2 (VGPR0), k0+3 (VGPR1), row/col = lane-16
// C/D 16x16 f32: element (vgpr i, lane L) = row i + 8*(L>=16), col L%16.
// ---------------------------------------------------------------------------
__global__ __launch_bounds__(256) void gemm_inter_wmma(
    const float* __restrict__ q,      // (M, QD) — q[:, :, 0, :] is contiguous
    const float* __restrict__ W,      // (QD, D) row-major
    float* __restrict__ inter)        // (M, D)
{
    const int tid  = threadIdx.x;
    const int wave = tid >> 5;
    const int lane = tid & 31;
    const int half = lane >> 4;       // 0: K pair {0,1}, 1: K pair {2,3}
    const int lr   = lane & 15;       // row (A) / col (B) within tile

    const int m0 = blockIdx.x * 16;   // row strip
    const int n0 = wave * 64;         // this wave's column strip

    v8f acc0 = {}, acc1 = {}, acc2 = {}, acc3 = {};

    const float* arow = q + (size_t)(m0 + lr) * QQD + 2 * half;

    #pragma unroll 4
    for (int k0 = 0; k0 < QQD; k0 += 4) {
        // A fragment: 2 consecutive fp32 per lane (64-bit load)
        v2f a;
        a.x = arow[k0 + 0];
        a.y = arow[k0 + 1];

        const float* brow0 = W + (size_t)(k0 + 2 * half) * DD;      // K row for VGPR0
        const float* brow1 = brow0 + DD;                            // K row for VGPR1

        v2f b0, b1, b2, b3;
        b0.x = brow0[n0 + 0 * 16 + lr];  b0.y = brow1[n0 + 0 * 16 + lr];
        b1.x = brow0[n0 + 1 * 16 + lr];  b1.y = brow1[n0 + 1 * 16 + lr];
        b2.x = brow0[n0 + 2 * 16 + lr];  b2.y = brow1[n0 + 2 * 16 + lr];
        b3.x = brow0[n0 + 3 * 16 + lr];  b3.y = brow1[n0 + 3 * 16 + lr];

        acc0 = __builtin_amdgcn_wmma_f32_16x16x4_f32(false, a, false, b0, (short)0, acc0, false, false);
        acc1 = __builtin_amdgcn_wmma_f32_16x16x4_f32(false, a, false, b1, (short)0, acc1, false, false);
        acc2 = __builtin_amdgcn_wmma_f32_16x16x4_f32(false, a, false, b2, (short)0, acc2, false, false);
        acc3 = __builtin_amdgcn_wmma_f32_16x16x4_f32(false, a, false, b3, (short)0, acc3, false, false);
    }

    // Store: row = i + 8*half, col = n0 + j*16 + lr
    const int rbase = m0 + 8 * half;
    #pragma unroll
    for (int i = 0; i < 8; ++i) {
        float* orow = inter + (size_t)(rbase + i) * DD + n0 + lr;
        orow[0 * 16] = acc0[i];
        orow[1 * 16] = acc1[i];
        orow[2 * 16] = acc2[i];
        orow[3 * 16] = acc3[i];
    }
}

// ---------------------------------------------------------------------------
// Kernel 2: per (b,n) fused epilogue. 2048 workgroups x 256 threads (8 waves).
//   T[a] = inter . t_rep[b,n,a,:]   (streams 134 MB once: b128 NT loads — BW bound)
//   G[o] = inter . obj_reps[b,o,:]  (L2-resident across the 64 n's of each b)
//   logits[a] = sum_o att1[a,o] G[o] + T[a] + bias
//   s = softmax(logits * m / t) * m ; att2 = s / (sum s + 1e-13)
//   out[o] = sum_a att2[a] att1[a,o]
// ---------------------------------------------------------------------------
__device__ __forceinline__ float wave_reduce_add(float v) {
    #pragma unroll
    for (int off = 16; off >= 1; off >>= 1) v += __shfl_xor(v, off, 32);
    return v;
}
__device__ __forceinline__ float wave_reduce_max(float v) {
    #pragma unroll
    for (int off = 16; off >= 1; off >>= 1) v = fmaxf(v, __shfl_xor(v, off, 32));
    return v;
}

__global__ __launch_bounds__(256) void fused_att(
    const float* __restrict__ inter,     // (B*N, D)
    const float* __restrict__ att1,      // (B, N, A, O)
    const float* __restrict__ obj_reps,  // (B, O, D)
    const int*   __restrict__ tags,      // (B, N, A)
    const float* __restrict__ t_rep,     // (B, N, A, D)
    const float* __restrict__ bias,      // scalar
    const int*   __restrict__ t_scalar,  // scalar int
    float* __restrict__ out)             // (B, N, O)
{
    __shared__ __align__(16) float inter_s[DD];
    __shared__ __align__(16) float att1_s[AA * OO];
    __shared__ float T_s[AA];
    __shared__ float G_s[OO];

    const int bn   = blockIdx.x;          // b*N + n
    const int b    = bn / NN;
    const int tid  = threadIdx.x;
    const int wave = tid >> 5;
    const int lane = tid & 31;

    const float* interRow = inter    + (size_t)bn * DD;
    const float* trepRow  = t_rep    + (size_t)bn * AA * DD;
    const float* att1Row  = att1     + (size_t)bn * AA * OO;
    const float* objB     = obj_reps + (size_t)b  * OO * DD;

    // Stage inter row (2 KB) and att1 tile (4 KB) into LDS with b128 ops.
    if (tid < DD / 4)
        ((v4f*)inter_s)[tid] = ((const v4f*)interRow)[tid];         // 128 lanes
    ((v4f*)att1_s)[tid] = ((const v4f*)att1Row)[tid];               // 256 = AA*OO/4
    __syncthreads();

    // T[a]: one wave per row a, 4 rows per wave; 512 B per wave per step,
    // non-temporal (t_rep is single-use; keep it out of L2).
    #pragma unroll
    for (int r = 0; r < 4; ++r) {
        const int a = wave + r * 8;
        const v4f* tp4 = (const v4f*)(trepRow + (size_t)a * DD);
        float s = 0.0f;
        #pragma unroll
        for (int i = 0; i < DD / 128; ++i) {                        // 4 x b128 per lane
            const v4f v = __builtin_nontemporal_load(tp4 + lane + 32 * i);
            const v4f w = ((const v4f*)inter_s)[lane + 32 * i];
            s += v.x * w.x + v.y * w.y + v.z * w.z + v.w * w.w;
        }
        s = wave_reduce_add(s);
        if (lane == 0) T_s[a] = s;
    }

    // G[o]: same pattern over obj_reps (regular temporal: hot in L2 across n).
    #pragma unroll
    for (int r = 0; r < 4; ++r) {
        const int o = wave + r * 8;
        const v4f* op4 = (const v4f*)(objB + (size_t)o * DD);
        float s = 0.0f;
        #pragma unroll
        for (int i = 0; i < DD / 128; ++i) {
            const v4f v = op4[lane + 32 * i];
            const v4f w = ((const v4f*)inter_s)[lane + 32 * i];
            s += v.x * w.x + v.y * w.y + v.z * w.z + v.w * w.w;
        }
        s = wave_reduce_add(s);
        if (lane == 0) G_s[o] = s;
    }
    __syncthreads();

    // Wave 0: logits -> masked softmax -> renormalize -> output matvec.
    if (tid < 32) {
        const int a = lane;
        float lg = T_s[a] + bias[0];
        #pragma unroll 8
        for (int o = 0; o < OO; ++o) lg += att1_s[a * OO + o] * G_s[o];

        const float mk   = (float)tags[(size_t)bn * AA + a];
        const float invt = 1.0f / (float)t_scalar[0];
        const float x    = lg * invt * mk;       // masked entries -> 0, still in softmax

        const float xmax = wave_reduce_max(x);
        const float e    = __expf(x - xmax);
        const float esum = wave_reduce_add(e);
        const float s    = (e / esum) * mk;
        const float ssum = wave_reduce_add(s);
        const float att2 = s / (ssum + 1e-13f);

        // out[o] = sum_a att2[a] * att1[a, o]; broadcast att2 via wave shuffles.
        float acc = 0.0f;
        #pragma unroll 8
        for (int ai = 0; ai < AA; ++ai) {
            const float wgt = __shfl(att2, ai, 32);
            acc += wgt * att1_s[ai * OO + lane];
        }
        out[(size_t)bn * OO + lane] = acc;
    }
}

extern "C" void kernel_launch(void* const* d_in, const int* in_sizes, int n_in,
                              void* d_out, int out_size, void* d_ws, size_t ws_size,
                              hipStream_t stream) {
    const float* q        = (const float*)d_in[0];  // (B, N, 1, QD)
    const float* att1     = (const float*)d_in[1];  // (B, N, A, O)
    const float* obj_reps = (const float*)d_in[2];  // (B, O, D)
    const int*   tags     = (const int*)  d_in[3];  // (B, N, A)
    const float* t_rep    = (const float*)d_in[4];  // (B, N, A, D)
    const float* W        = (const float*)d_in[5];  // (QD, D)
    const float* bias     = (const float*)d_in[6];  // scalar
    const int*   t        = (const int*)  d_in[7];  // scalar
    float*       out      = (float*)d_out;          // (B, N, O)

    float* inter = (float*)d_ws;                    // (B*N, D) = 4 MB scratch

    gemm_inter_wmma<<<dim3(MM / 16), dim3(256), 0, stream>>>(q, W, inter);
    fused_att<<<dim3(MM), dim3(256), 0, stream>>>(inter, att1, obj_reps, tags,
                                                  t_rep, bias, t, out);
}